// GeneralOrderingRepair_20615843021332
// MI455X (gfx1250) — compile-verified
//
#include <hip/hip_runtime.h>
#include <hip/hip_bf16.h>

typedef __attribute__((ext_vector_type(2))) float v2f;
typedef __attribute__((ext_vector_type(8))) float v8f;

#define PP 8
#define DD 2
#define CC 64
#define TT 16

// ---------------------------------------------------------------------------
// Kernel 0: pack each of the 16 post-graphs into 64-bit row (successor) and
// column (predecessor) bitmasks.  rowmask[t][i] bit j = g[t][i][j].
// colmask[t][j] bit i = g[t][i][j].
// ---------------------------------------------------------------------------
__global__ __launch_bounds__(256) void prep_kernel(
    const int* __restrict__ g,
    unsigned long long* __restrict__ rowmask,
    unsigned long long* __restrict__ colmask)
{
    int tid = blockIdx.x * blockDim.x + threadIdx.x;
    if (tid >= TT * CC) return;
    int t = tid >> 6;
    int a = tid & 63;
    unsigned long long rm = 0ull, cm = 0ull;
    for (int k = 0; k < CC; ++k) {
        if (g[t * CC * CC + a * CC + k]) rm |= (1ull << k);
        if (g[t * CC * CC + k * CC + a]) cm |= (1ull << k);
    }
    rowmask[tid] = rm;
    colmask[tid] = cm;
}

// ---------------------------------------------------------------------------
// Kernel 1: viol[b,t] = sum_{i,j} relu(y[b,j]-y[b,i]) * g[t][i][j]
// via V_WMMA_F32_16X16X4_F32.  One wave per 16-row batch tile; A (16x4 f32)
// is the on-the-fly relu-diff tile, B (4x16 f32) is the graph bit tile.
// A layout (ISA 7.12.2): lanes 0-15 -> M, VGPR0=K(0|2), VGPR1=K(1|3) by half.
// C/D layout: lane<16 -> M=r, lane>=16 -> M=r+8; N = lane&15.
// ---------------------------------------------------------------------------
__global__ __launch_bounds__(32) void viol_wmma_kernel(
    const float* __restrict__ y,
    const unsigned long long* __restrict__ rowmask,
    float* __restrict__ viol)
{
    __shared__ float ytile[16][CC];
    __shared__ unsigned long long gm[TT][CC];

    const int lane = threadIdx.x;          // 0..31, full wave
    const int b0 = blockIdx.x * 16;

    for (int idx = lane; idx < 16 * CC; idx += 32)
        ytile[idx >> 6][idx & 63] = y[(b0 + (idx >> 6)) * CC + (idx & 63)];
    for (int idx = lane; idx < TT * CC; idx += 32)
        gm[idx >> 6][idx & 63] = rowmask[idx];
    __syncthreads();

    const int M  = lane & 15;
    const int hi = lane >> 4;              // 0: K even base, 1: K+2
    const int t  = lane & 15;

    v8f acc = {};
    for (int i = 0; i < CC; ++i) {
        const float yi = ytile[M][i];
        const unsigned long long m = gm[t][i];
        for (int jb = 0; jb < CC; jb += 4) {
            const int j0 = jb + 2 * hi;
            float a0 = fmaxf(ytile[M][j0]     - yi, 0.0f);
            float a1 = fmaxf(ytile[M][j0 + 1] - yi, 0.0f);
            v2f a = {a0, a1};
            v2f b = {(float)((m >> j0) & 1ull), (float)((m >> (j0 + 1)) & 1ull)};
            acc = __builtin_amdgcn_wmma_f32_16x16x4_f32(
                false, a, false, b, (short)0, acc, false, false);
        }
    }

    #pragma unroll
    for (int r = 0; r < 8; ++r)
        viol[(b0 + r + 8 * hi) * TT + t] = acc[r];
}

// ---------------------------------------------------------------------------
// Kernel 2: per-row repair.  One block (64 threads = node j) per batch row.
// ---------------------------------------------------------------------------
__global__ __launch_bounds__(64) void repair_kernel(
    const float* __restrict__ x,
    const float* __restrict__ y,
    const float* __restrict__ viol,
    const unsigned long long* __restrict__ colmask,
    float* __restrict__ out)
{
    __shared__ float ys[CC];
    __shared__ float yd[CC];
    __shared__ float yr[CC];
    __shared__ int   sel[PP];
    __shared__ unsigned long long wkey[2];
    __shared__ float wsum[2];
    __shared__ int   badflag;

    const int b = blockIdx.x;
    const int j = threadIdx.x;
    const float yj = y[b * CC + j];
    ys[j] = yj;
    if (j == 0) badflag = 0;
    if (j < PP) {
        float v0 = viol[b * TT + 2 * j];
        float v1 = viol[b * TT + 2 * j + 1];
        int d = (v1 < v0) ? 1 : 0;                       // stable argsort -> tie picks 0
        sel[j] = (x[b * CC + j] > 0.0f) ? (2 * j + d) : -1;
    }
    __syncthreads();

    // predecessor bitmask of node j for this row's merged graph
    unsigned long long pred = 0ull;
    #pragma unroll
    for (int p = 0; p < PP; ++p) {
        int s = sel[p];
        if (s >= 0) pred |= colmask[s * CC + j];
    }

    // sat: every edge i->j must have y[i] >= y[j]
    {
        unsigned long long m = pred;
        bool bad = false;
        while (m) {
            int i = __builtin_ctzll(m);
            m &= m - 1;
            if (ys[i] < yj) bad = true;
        }
        if (bad) atomicOr(&badflag, 1);
    }

    // topological ranks: 64 rounds of "argmax y among zero-indegree remaining"
    unsigned long long rem = ~0ull;
    int myrank = 0;
    for (int r = 0; r < CC; ++r) {
        unsigned long long key = 0ull;
        bool avail = ((rem >> j) & 1ull) && ((pred & rem) == 0ull);
        if (avail) {
            unsigned int u = __float_as_uint(yj);
            unsigned int k32 = (u & 0x80000000u) ? ~u : (u | 0x80000000u);
            key = ((unsigned long long)k32 << 6) | (unsigned long long)(63 - j);
        }
        #pragma unroll
        for (int off = 16; off >= 1; off >>= 1) {
            unsigned long long o = __shfl_xor(key, off, 32);
            if (o > key) key = o;
        }
        if ((j & 31) == 0) wkey[j >> 5] = key;
        __syncthreads();
        unsigned long long mk = wkey[0] > wkey[1] ? wkey[0] : wkey[1];
        int idx = (mk == 0ull) ? 0 : (63 - (int)(mk & 63ull));  // tie -> smallest j
        if (j == idx) myrank = r;
        rem &= ~(1ull << idx);
        __syncthreads();
    }

    // stable descending sort of y via rank-by-count
    int pos = 0;
    for (int i = 0; i < CC; ++i) {
        float yi = ys[i];
        if (yi > yj || (yi == yj && i < j)) pos++;
    }
    yd[pos] = yj;
    __syncthreads();

    const bool sat = (badflag == 0);
    const float yrep = sat ? yj : yd[myrank];
    yr[j] = yrep;
    out[b * (CC + 1) + j] = yrep;
    __syncthreads();

    // margin = sum over edges of relu(yrep[j] - yrep[i])
    float msum = 0.0f;
    {
        unsigned long long m = pred;
        while (m) {
            int i = __builtin_ctzll(m);
            m &= m - 1;
            float d = yrep - yr[i];
            if (d > 0.0f) msum += d;
        }
    }
    #pragma unroll
    for (int off = 16; off >= 1; off >>= 1) msum += __shfl_xor(msum, off, 32);
    if ((j & 31) == 0) wsum[j >> 5] = msum;
    __syncthreads();
    if (j == 0) {
        float margin = wsum[0] + wsum[1];
        float bot = (margin > 0.0f) ? (yd[0] + margin) : -__builtin_inff();
        out[b * (CC + 1) + CC] = bot;
    }
}

// ---------------------------------------------------------------------------
extern "C" void kernel_launch(void* const* d_in, const int* in_sizes, int n_in,
                              void* d_out, int out_size, void* d_ws, size_t ws_size,
                              hipStream_t stream) {
    const float* x  = (const float*)d_in[0];
    const float* y  = (const float*)d_in[1];
    const int*   pg = (const int*)d_in[2];
    float* out = (float*)d_out;

    const int B = in_sizes[1] / CC;   // y is (B, 64)

    unsigned long long* rowmask = (unsigned long long*)d_ws;
    unsigned long long* colmask = rowmask + TT * CC;
    float* viol = (float*)(colmask + TT * CC);

    prep_kernel<<<(TT * CC + 255) / 256, 256, 0, stream>>>(pg, rowmask, colmask);
    viol_wmma_kernel<<<B / 16, 32, 0, stream>>>(y, rowmask, viol);
    repair_kernel<<<B, 64, 0, stream>>>(x, y, viol, colmask, out);
}